// LSTM_Layer_v0_12223476924616
// MI455X (gfx1250) — compile-verified
//
#include <hip/hip_runtime.h>
#include <math.h>

// LSTM layer: T=512, B=32, I=H=1024, fp32, CDNA5 (gfx1250) WMMA f32 path.
//
// Two-phase (when ws_size allows):
//   Phase 1 (parallel): xU[g] = x @ U_g + b_g -> d_ws. M-blocked 8x: each wave
//     holds 8 accumulators -> 8 WMMAs per B-operand load (U traffic /8).
//   Phase 2 (serial):   per step t: preact = xU[g][t] + h_{t-1} @ V_g (K=1024).
//     One block owns all 32 batch rows: 2 WMMAs per B load (V traffic /2),
//     fused sigmoid/tanh c/h update. V (16MB) stays resident in the 192MB L2.
// Fallback (small ws): fused K=2048 step kernel ([x_t | h_{t-1}] @ [U;V]).

#define T_SEQ 512
#define BATCH 32
#define ISZ   1024
#define HSZ   1024
#define KC    128            // K-chunk staged in LDS
#define MT1   8              // M-tiles per block in phase 1 (128 rows)

typedef __attribute__((ext_vector_type(2))) float v2f;
typedef __attribute__((ext_vector_type(8))) float v8f;

__device__ __forceinline__ float sigmoid_(float x) { return 1.0f / (1.0f + __expf(-x)); }

__global__ void lstm_init_kernel(float* __restrict__ h0, float* __restrict__ cbuf) {
    int i = blockIdx.x * blockDim.x + threadIdx.x;
    if (i < BATCH * HSZ) { h0[i] = 0.0f; cbuf[i] = 0.0f; }
}

// ---------------------------------------------------------------------------
// Phase 1: xU[g][row][n] = sum_k x[row][k] * U_g[k][n] + b_g[n]
// grid = (HSZ/16, (T*B)/(16*MT1)), block = 128 (4 waves = 4 gates)
// Each wave: 8 M-tiles -> 8 WMMAs per B-pair load.
// ---------------------------------------------------------------------------
__global__ __launch_bounds__(128)
void lstm_xu_gemm_kernel(const float* __restrict__ x,
                         const float* __restrict__ Ui, const float* __restrict__ bi,
                         const float* __restrict__ Uf, const float* __restrict__ bf,
                         const float* __restrict__ Uc, const float* __restrict__ bc,
                         const float* __restrict__ Uo, const float* __restrict__ bo,
                         float* __restrict__ xu)      // [4][T*B][HSZ]
{
    __shared__ float As[16 * MT1 * KC];   // 128 rows x 128 K = 64 KB

    const int tid   = threadIdx.x;
    const int wave  = tid >> 5;
    const int lane  = tid & 31;
    const int ntile = blockIdx.x;         // 0..63
    const int mgrp  = blockIdx.y;         // 0..127 (128 rows each)

    const float* Ug = (wave == 0) ? Ui : (wave == 1) ? Uf : (wave == 2) ? Uc : Uo;
    const float* bg = (wave == 0) ? bi : (wave == 1) ? bf : (wave == 2) ? bc : bo;
    float* outg = xu + (size_t)wave * (T_SEQ * BATCH) * HSZ;

    const int n     = ntile * 16 + (lane & 15);
    const int koff2 = (lane >> 4) << 1;
    const int mA    = lane & 15;

    v8f acc[MT1];
    #pragma unroll
    for (int mt = 0; mt < MT1; ++mt) acc[mt] = (v8f){};

    for (int kc = 0; kc < ISZ; kc += KC) {
        // cooperative load: 128 rows x KC floats = 4096 float4, 32/thread
        const float* srcbase = x + (size_t)mgrp * (16 * MT1) * ISZ + kc;
        #pragma unroll 4
        for (int j = 0; j < 32; ++j) {
            int f  = tid + 128 * j;
            int m  = f >> 5;                  // KC/4 = 32 float4 per row
            int kv = (f & 31) << 2;
            *reinterpret_cast<float4*>(&As[m * KC + kv]) =
                *reinterpret_cast<const float4*>(srcbase + (size_t)m * ISZ + kv);
        }
        __syncthreads();

        const float* Bp = Ug + (size_t)kc * HSZ;
        #pragma unroll 2
        for (int kk = 0; kk < KC; kk += 4) {
            v2f b;
            b.x = Bp[(size_t)(kk + koff2)     * HSZ + n];
            b.y = Bp[(size_t)(kk + koff2 + 1) * HSZ + n];
            #pragma unroll
            for (int mt = 0; mt < MT1; ++mt) {
                v2f a = *reinterpret_cast<const v2f*>(
                    &As[(mt * 16 + mA) * KC + kk + koff2]);
                acc[mt] = __builtin_amdgcn_wmma_f32_16x16x4_f32(
                    false, a, false, b, (short)0, acc[mt], false, false);
            }
        }
        __syncthreads();
    }

    const float bn    = bg[n];
    const int   mbase = (lane >> 4) << 3;
    #pragma unroll
    for (int mt = 0; mt < MT1; ++mt)
        #pragma unroll
        for (int r = 0; r < 8; ++r)
            outg[(size_t)(mgrp * (16 * MT1) + mt * 16 + mbase + r) * HSZ + n] =
                acc[mt][r] + bn;
}

// ---------------------------------------------------------------------------
// Phase 2 step: preact = xU[g][t] + h_{t-1} @ V_g, K=1024.
// grid = (HSZ/16), block = 128 (4 waves = 4 gates); each wave owns both
// 16-row M-tiles (all 32 batch rows) -> 2 WMMAs per B-pair load.
// ---------------------------------------------------------------------------
__global__ __launch_bounds__(128)
void lstm_step2_kernel(const float* __restrict__ Vi, const float* __restrict__ Vf,
                       const float* __restrict__ Vc, const float* __restrict__ Vo,
                       const float* __restrict__ xu,   // [4][T*B][HSZ]
                       const float* __restrict__ h0,
                       float* __restrict__ hseq,       // [T, B, H]
                       float* __restrict__ cbuf,       // running c [B, H]
                       int t)
{
    __shared__ float As[32 * KC];        // 32 rows x 128 K = 16 KB
    __shared__ float gbuf[4][32][16];    // per-gate 32x16 preactivations (8 KB)

    const int tid   = threadIdx.x;
    const int wave  = tid >> 5;
    const int lane  = tid & 31;
    const int ntile = blockIdx.x;

    const float* hprev = (t == 0) ? h0 : (hseq + (size_t)(t - 1) * BATCH * HSZ);
    const float* Vg  = (wave == 0) ? Vi : (wave == 1) ? Vf : (wave == 2) ? Vc : Vo;
    const float* xug = xu + (size_t)wave * (T_SEQ * BATCH) * HSZ;

    const int n     = ntile * 16 + (lane & 15);
    const int koff2 = (lane >> 4) << 1;
    const int mA    = lane & 15;

    v8f acc0 = {}, acc1 = {};

    for (int kc = 0; kc < HSZ; kc += KC) {
        // cooperative load: 32 rows x KC floats = 1024 float4, 8/thread
        const float* srcbase = hprev + kc;
        #pragma unroll
        for (int j = 0; j < 8; ++j) {
            int f  = tid + 128 * j;
            int m  = f >> 5;
            int kv = (f & 31) << 2;
            *reinterpret_cast<float4*>(&As[m * KC + kv]) =
                *reinterpret_cast<const float4*>(srcbase + (size_t)m * HSZ + kv);
        }
        __syncthreads();

        const float* Bp = Vg + (size_t)kc * HSZ;
        if (kc + KC < HSZ)
            __builtin_prefetch(Bp + (size_t)(KC + (lane & 15) * 8) * HSZ + n, 0, 0);
        #pragma unroll 8
        for (int kk = 0; kk < KC; kk += 4) {
            v2f b;
            b.x = Bp[(size_t)(kk + koff2)     * HSZ + n];
            b.y = Bp[(size_t)(kk + koff2 + 1) * HSZ + n];
            v2f a0 = *reinterpret_cast<const v2f*>(&As[mA        * KC + kk + koff2]);
            v2f a1 = *reinterpret_cast<const v2f*>(&As[(16 + mA) * KC + kk + koff2]);
            acc0 = __builtin_amdgcn_wmma_f32_16x16x4_f32(
                false, a0, false, b, (short)0, acc0, false, false);
            acc1 = __builtin_amdgcn_wmma_f32_16x16x4_f32(
                false, a1, false, b, (short)0, acc1, false, false);
        }
        __syncthreads();
    }

    // add precomputed x@U + b term and exchange gate tiles through LDS
    const int mbase = (lane >> 4) << 3;
    #pragma unroll
    for (int r = 0; r < 8; ++r) {
        size_t row0 = (size_t)t * BATCH + mbase + r;
        gbuf[wave][mbase + r][lane & 15]      = acc0[r] + xug[row0 * HSZ + n];
        gbuf[wave][16 + mbase + r][lane & 15] = acc1[r] + xug[(row0 + 16) * HSZ + n];
    }
    __syncthreads();

    // fused pointwise: 32x16 = 512 elements over 128 threads
    #pragma unroll
    for (int e = tid; e < 512; e += 128) {
        int M   = e >> 4;                 // batch row 0..31
        int nl  = e & 15;
        int col = ntile * 16 + nl;
        float it = sigmoid_(gbuf[0][M][nl]);
        float ft = sigmoid_(gbuf[1][M][nl]);
        float gt = tanhf  (gbuf[2][M][nl]);
        float ot = sigmoid_(gbuf[3][M][nl]);
        size_t idx = (size_t)M * HSZ + col;
        float c = ft * cbuf[idx] + it * gt;
        float h = ot * tanhf(c);
        cbuf[idx] = c;
        hseq[(size_t)t * BATCH * HSZ + idx] = h;
    }
}

// ---------------------------------------------------------------------------
// Fallback fused step (small workspace): K = 2048 over [x_t | h_{t-1}]
// grid = (HSZ/16, BATCH/16), block = 128
// ---------------------------------------------------------------------------
__global__ __launch_bounds__(128)
void lstm_step_fused_kernel(const float* __restrict__ x,
                            const float* __restrict__ Ui, const float* __restrict__ Vi, const float* __restrict__ bi,
                            const float* __restrict__ Uf, const float* __restrict__ Vf, const float* __restrict__ bf,
                            const float* __restrict__ Uc, const float* __restrict__ Vc, const float* __restrict__ bc,
                            const float* __restrict__ Uo, const float* __restrict__ Vo, const float* __restrict__ bo,
                            const float* __restrict__ h0,
                            float* __restrict__ hseq, float* __restrict__ cbuf, int t)
{
    __shared__ float As[16 * KC];
    __shared__ float gbuf[4][16][16];

    const int tid   = threadIdx.x;
    const int wave  = tid >> 5;
    const int lane  = tid & 31;
    const int ntile = blockIdx.x;
    const int mtile = blockIdx.y;

    const float* hprev = (t == 0) ? h0 : (hseq + (size_t)(t - 1) * BATCH * HSZ);
    const float* xt    = x + (size_t)t * BATCH * ISZ;

    const float* Ug = (wave == 0) ? Ui : (wave == 1) ? Uf : (wave == 2) ? Uc : Uo;
    const float* Vg = (wave == 0) ? Vi : (wave == 1) ? Vf : (wave == 2) ? Vc : Vo;
    const float* bg = (wave == 0) ? bi : (wave == 1) ? bf : (wave == 2) ? bc : bo;

    const int n     = ntile * 16 + (lane & 15);
    const int koff2 = (lane >> 4) << 1;
    const int mA    = lane & 15;

    v8f acc = {};

    for (int kc = 0; kc < 2 * HSZ; kc += KC) {
        const bool   fromX   = (kc < ISZ);
        const float* srcbase = fromX ? (xt + kc) : (hprev + (kc - ISZ));
        #pragma unroll
        for (int j = 0; j < 4; ++j) {
            int f  = tid + 128 * j;
            int m  = f >> 5;
            int kv = (f & 31) << 2;
            *reinterpret_cast<float4*>(&As[m * KC + kv]) =
                *reinterpret_cast<const float4*>(srcbase + (size_t)(mtile * 16 + m) * ISZ + kv);
        }
        __syncthreads();

        const float* Bp = fromX ? (Ug + (size_t)kc * HSZ) : (Vg + (size_t)(kc - ISZ) * HSZ);
        #pragma unroll 8
        for (int kk = 0; kk < KC; kk += 4) {
            v2f a = *reinterpret_cast<const v2f*>(&As[mA * KC + kk + koff2]);
            v2f b;
            b.x = Bp[(size_t)(kk + koff2)     * HSZ + n];
            b.y = Bp[(size_t)(kk + koff2 + 1) * HSZ + n];
            acc = __builtin_amdgcn_wmma_f32_16x16x4_f32(
                false, a, false, b, (short)0, acc, false, false);
        }
        __syncthreads();
    }

    const float bn    = bg[n];
    const int   mbase = (lane >> 4) << 3;
    #pragma unroll
    for (int r = 0; r < 8; ++r) gbuf[wave][mbase + r][lane & 15] = acc[r] + bn;
    __syncthreads();

    #pragma unroll
    for (int e = tid; e < 256; e += 128) {
        int M   = e >> 4;
        int nl  = e & 15;
        int row = mtile * 16 + M;
        int col = ntile * 16 + nl;
        float it = sigmoid_(gbuf[0][M][nl]);
        float ft = sigmoid_(gbuf[1][M][nl]);
        float gt = tanhf  (gbuf[2][M][nl]);
        float ot = sigmoid_(gbuf[3][M][nl]);
        size_t idx = (size_t)row * HSZ + col;
        float c = ft * cbuf[idx] + it * gt;
        float h = ot * tanhf(c);
        cbuf[idx] = c;
        hseq[(size_t)t * BATCH * HSZ + idx] = h;
    }
}

__global__ void lstm_finalize_kernel(const float* __restrict__ hseq, float* __restrict__ hT) {
    int i = blockIdx.x * blockDim.x + threadIdx.x;
    if (i < BATCH * HSZ) hT[i] = hseq[(size_t)(T_SEQ - 1) * BATCH * HSZ + i];
}

extern "C" void kernel_launch(void* const* d_in, const int* in_sizes, int n_in,
                              void* d_out, int out_size, void* d_ws, size_t ws_size,
                              hipStream_t stream) {
    (void)in_sizes; (void)n_in; (void)out_size;

    const float* x  = (const float*)d_in[0];
    const float* Ui = (const float*)d_in[1];
    const float* Vi = (const float*)d_in[2];
    const float* bi = (const float*)d_in[3];
    const float* Uf = (const float*)d_in[4];
    const float* Vf = (const float*)d_in[5];
    const float* bf = (const float*)d_in[6];
    const float* Uc = (const float*)d_in[7];
    const float* Vc = (const float*)d_in[8];
    const float* bc = (const float*)d_in[9];
    const float* Uo = (const float*)d_in[10];
    const float* Vo = (const float*)d_in[11];
    const float* bo = (const float*)d_in[12];

    float* out  = (float*)d_out;
    float* hseq = out;                                    // [T, B, H]
    float* hT   = out + (size_t)T_SEQ * BATCH * HSZ;      // [B, H]
    float* cT   = hT + (size_t)BATCH * HSZ;               // [B, H] (running c)
    float* h0   = (float*)d_ws;                           // zero h_0 (first 128 KB)
    float* xu   = h0 + (size_t)BATCH * HSZ;               // [4][T*B][H] when it fits

    const size_t need = ((size_t)BATCH * HSZ + (size_t)4 * T_SEQ * BATCH * HSZ) * sizeof(float);

    lstm_init_kernel<<<(BATCH * HSZ + 255) / 256, 256, 0, stream>>>(h0, cT);

    if (ws_size >= need) {
        // Phase 1: fully parallel x@U + b for all gates (M-blocked 8x).
        dim3 xuGrid(HSZ / 16, (T_SEQ * BATCH) / (16 * MT1));   // 64 x 128
        lstm_xu_gemm_kernel<<<xuGrid, 128, 0, stream>>>(
            x, Ui, bi, Uf, bf, Uc, bc, Uo, bo, xu);
        // Phase 2: serial recurrence, K=1024 (h@V only), one block per N-tile.
        for (int t = 0; t < T_SEQ; ++t)
            lstm_step2_kernel<<<dim3(HSZ / 16), 128, 0, stream>>>(
                Vi, Vf, Vc, Vo, xu, h0, hseq, cT, t);
    } else {
        // Fallback: fused K=2048 steps, no large scratch needed.
        dim3 stepGrid(HSZ / 16, BATCH / 16);
        for (int t = 0; t < T_SEQ; ++t)
            lstm_step_fused_kernel<<<stepGrid, 128, 0, stream>>>(
                x, Ui, Vi, bi, Uf, Vf, bf, Uc, Vc, bc, Uo, Vo, bo,
                h0, hseq, cT, t);
    }

    lstm_finalize_kernel<<<(BATCH * HSZ + 255) / 256, 256, 0, stream>>>(hseq, hT);
}